// BondLenConstrain_54004918780082
// MI455X (gfx1250) — compile-verified
//
#include <hip/hip_runtime.h>
#include <hip/hip_bf16.h>
#include <math.h>

#define B_    16
#define CH_   8
#define R_    8192
#define NALT_ 10
#define EPS_  1e-10f
#define DEG_  57.29577951308232f   // 180/pi
#define TWO_PI_ 6.283185307179586f

typedef __attribute__((address_space(1))) int g_int_t;   // global
typedef __attribute__((address_space(3))) int l_int_t;   // LDS

// ---------------------------------------------------------------------------
// small float3 helpers
// ---------------------------------------------------------------------------
struct V3 { float x, y, z; };
__device__ __forceinline__ V3 ld3(const float* p) { return {p[0], p[1], p[2]}; }
__device__ __forceinline__ V3 vsub(V3 a, V3 b) { return {a.x - b.x, a.y - b.y, a.z - b.z}; }
__device__ __forceinline__ float vdot(V3 a, V3 b) { return a.x * b.x + a.y * b.y + a.z * b.z; }

__device__ __forceinline__ float angle_deg(V3 a, V3 b, bool* mask) {
    float na = sqrtf(vdot(a, a));
    float nb = sqrtf(vdot(b, b));
    *mask = (na > 0.0f) && (nb > 0.0f);
    float c = vdot(a, b) / fmaxf(na * nb, 1e-12f);
    c = fminf(1.0f, fmaxf(-1.0f, c));
    return acosf(c) * DEG_;
}

// ---------------------------------------------------------------------------
// Kernel 1: init lookup table to -1 (16B stores; count divisible by 4)
// ---------------------------------------------------------------------------
__global__ void bl_init_table(int4* __restrict__ table4, int n4) {
    int i = blockIdx.x * blockDim.x + threadIdx.x;
    if (i < n4) table4[i] = make_int4(-1, -1, -1, -1);
}

// ---------------------------------------------------------------------------
// Kernel 2: scatter atom indices into table[b][ch][r][at]
// ---------------------------------------------------------------------------
__global__ void bl_scatter(const int* __restrict__ desc, int* __restrict__ table, int n_atoms) {
    int i = blockIdx.x * blockDim.x + threadIdx.x;
    if (i >= n_atoms) return;
    const int* row = desc + i * 5;
    int at = row[0], r = row[1], ch = row[2], b = row[3];
    if ((unsigned)at < 3u && (unsigned)r < (unsigned)R_ &&
        (unsigned)ch < (unsigned)CH_ && (unsigned)b < (unsigned)B_) {
        table[(((b * CH_) + ch) * R_ + r) * 3 + at] = i;
    }
}

// ---------------------------------------------------------------------------
// Kernel 3: per-residue-slot score + full output write (zeros included)
//   gid = (b*CH + ch)*R + r  ;  table base for this slot = gid*3
//   C  of res r   : table[gid*3 + 2]   (C_TYPE  = 2)
//   CA of res r   : table[gid*3 + 1]   (CA_TYPE = 1)
//   N  of res r+1 : table[gid*3 + 3]   (N_TYPE  = 0)
//   CA of res r+1 : table[gid*3 + 4]
// mean/std (20x3 each) staged into LDS via gfx1250 async-load-to-LDS.
// ---------------------------------------------------------------------------
__global__ void bl_score(const int* __restrict__ desc,
                         const float* __restrict__ coords,
                         const float* __restrict__ weight,
                         const float* __restrict__ mean,
                         const float* __restrict__ stdv,
                         const int* __restrict__ table,
                         float* __restrict__ out,
                         int slots) {
    __shared__ float s_tab[128];   // [0..59]=mean, [64..123]=std
    int t = threadIdx.x;

#if defined(__gfx1250__) && __has_builtin(__builtin_amdgcn_global_load_async_to_lds_b32)
    if (t < 124 && (t < 60 || t >= 64)) {
        const float* g = (t < 60) ? (mean + t) : (stdv + (t - 64));
        __builtin_amdgcn_global_load_async_to_lds_b32(
            (g_int_t*)g,
            (l_int_t*)&s_tab[t],
            0, 0);
    }
  #if __has_builtin(__builtin_amdgcn_s_wait_asynccnt)
    __builtin_amdgcn_s_wait_asynccnt(0);
  #else
    asm volatile("s_wait_asynccnt 0" ::: "memory");
  #endif
#else
    if (t < 60)                  s_tab[t] = mean[t];
    else if (t >= 64 && t < 124) s_tab[t] = stdv[t - 64];
#endif
    __syncthreads();

    int gid = blockIdx.x * blockDim.x + t;
    if (gid >= slots) return;

    int r = gid & (R_ - 1);          // R_ is a power of two
    float val = 0.0f;

    if (r < R_ - 1) {
        int tb   = gid * 3;
        int cacI = table[tb + 1];
        int cI   = table[tb + 2];
        int nI   = table[tb + 3];
        int canI = table[tb + 4];
        // all four >= 0  <=>  OR of sign bits is clear
        if ((cacI | cI | nI | canI) >= 0) {
            V3 cc   = ld3(coords + cI   * 3);
            V3 nc   = ld3(coords + nI   * 3);
            V3 cacc = ld3(coords + cacI * 3);
            V3 canc = ld3(coords + canI * 3);

            V3 vcn = vsub(nc, cc);                   // N - C
            float blen = sqrtf(vdot(vcn, vcn));

            bool m1, m2;
            float ang1 = angle_deg(vcn, vsub(canc, nc), &m1);          // C->N vs CA(+1)-N
            float ang2 = angle_deg(vsub(cc, cacc), vsub(cc, nc), &m2); // C-CA vs -(N-C)

            if (m1 && m2) {
                int seq = desc[cI * 5 + 4];
                seq = min(max(seq, 0), 19);          // JAX clamps OOB gathers

                float x[3] = {blen, ang1, ang2};
                float s = 0.0f;
#pragma unroll
                for (int k = 0; k < 3; ++k) {
                    float mu  = s_tab[seq * 3 + k];
                    float sd  = s_tab[64 + seq * 3 + k];
                    float var = sd * sd;
                    float denom = sqrtf(TWO_PI_ * var);
                    float d = x[k] - mu;
                    float pdf = __expf(-d * d / (2.0f * var)) / denom;
                    float ldn = __logf(denom);
                    s += -(__logf(fmaxf(pdf, EPS_)) + ldn);
                }
                float w = weight[0];
                val = s * (1.0f + tanhf(w));          // 1 - tanh(-w)
            }
        }
    }

    // Write all NALT slots: value in slot 0, zeros elsewhere.
    // byte offset = gid*40, always 8-aligned -> 5x float2 stores.
    float2* o = (float2*)(out + (size_t)gid * NALT_);
    o[0] = make_float2(val, 0.0f);
    o[1] = make_float2(0.0f, 0.0f);
    o[2] = make_float2(0.0f, 0.0f);
    o[3] = make_float2(0.0f, 0.0f);
    o[4] = make_float2(0.0f, 0.0f);
}

// ---------------------------------------------------------------------------
extern "C" void kernel_launch(void* const* d_in, const int* in_sizes, int n_in,
                              void* d_out, int out_size, void* d_ws, size_t ws_size,
                              hipStream_t stream) {
    const int*   desc   = (const int*)  d_in[0];   // atom_description (n,5) int32
    const float* coords = (const float*)d_in[1];   // (n,3) f32
    // d_in[2] = alternatives (unused by reference output)
    const float* weight = (const float*)d_in[3];   // (1,) f32
    const float* mean   = (const float*)d_in[4];   // (20,3) f32
    const float* stdv   = (const float*)d_in[5];   // (20,3) f32
    float* out = (float*)d_out;
    int*   table = (int*)d_ws;                     // B*CH*R*3 ints = 12.6 MB

    int n_atoms = in_sizes[0] / 5;
    int n_tab   = B_ * CH_ * R_ * 3;
    int n4      = n_tab / 4;
    int slots   = B_ * CH_ * R_;

    dim3 blk(256);
    bl_init_table<<<(n4 + 255) / 256, blk, 0, stream>>>((int4*)table, n4);
    bl_scatter  <<<(n_atoms + 255) / 256, blk, 0, stream>>>(desc, table, n_atoms);
    bl_score    <<<(slots + 255) / 256, blk, 0, stream>>>(desc, coords, weight, mean,
                                                          stdv, table, out, slots);
}